// MXEPLoss_27831388078896
// MI455X (gfx1250) — compile-verified
//
#include <hip/hip_runtime.h>
#include <hip/hip_bf16.h>
#include <math.h>

#define NS   4096
#define NQ   4096
#define DIM  1024
#define KCLS 128
#define EPS_ 1e-12f

typedef __attribute__((ext_vector_type(2))) float v2f;
typedef __attribute__((ext_vector_type(8))) float v8f;

// ---------------------------------------------------------------------------
// Kernel 0: zero the scalar output (harness poisons d_out with 0xAA).
// ---------------------------------------------------------------------------
__global__ void zero_out_kernel(float* out) { out[0] = 0.0f; }

// ---------------------------------------------------------------------------
// Kernel A: per-class statistics.  One block per class k:
//   mus[k,:]   = sum of xs rows with ys==k                (class sum)
//   protoT[:,k]= mus[k,:]/clip(count,0.1)  (transposed for WMMA B-feeding)
//   cnt[k], S2[k]=sum ||xs_j||^2 over class, mu2[k]=||mus_k||^2, pn2[k]=||proto_n_k||^2
// ---------------------------------------------------------------------------
__global__ void __launch_bounds__(256)
class_stats_kernel(const float* __restrict__ xs, const int* __restrict__ ys,
                   float* __restrict__ mus, float* __restrict__ protoT,
                   float* __restrict__ cls_cnt, float* __restrict__ cls_s2,
                   float* __restrict__ cls_mu2, float* __restrict__ cls_pn2)
{
    __shared__ int   s_ys[NS];
    __shared__ float s_red[256];
    const int k   = blockIdx.x;
    const int tid = threadIdx.x;
    for (int j = tid; j < NS; j += 256) s_ys[j] = ys[j];
    __syncthreads();

    const int d0 = tid * 4;                 // 256 threads * 4 dims = DIM
    float a0 = 0.f, a1 = 0.f, a2 = 0.f, a3 = 0.f;
    float s2 = 0.f;
    int   cnt = 0;
    for (int j = 0; j < NS; ++j) {
        if (s_ys[j] == k) {
            const float* row = xs + j * DIM + d0;
            float x0 = row[0], x1 = row[1], x2 = row[2], x3 = row[3];
            a0 += x0; a1 += x1; a2 += x2; a3 += x3;
            s2 += x0 * x0 + x1 * x1 + x2 * x2 + x3 * x3;
            ++cnt;
        }
    }
    const float cntf  = (float)cnt;
    const float clipc = fmaxf(cntf, 0.1f);
    const float inv   = 1.0f / clipc;

    mus[k * DIM + d0 + 0] = a0;
    mus[k * DIM + d0 + 1] = a1;
    mus[k * DIM + d0 + 2] = a2;
    mus[k * DIM + d0 + 3] = a3;
    protoT[(d0 + 0) * KCLS + k] = a0 * inv;
    protoT[(d0 + 1) * KCLS + k] = a1 * inv;
    protoT[(d0 + 2) * KCLS + k] = a2 * inv;
    protoT[(d0 + 3) * KCLS + k] = a3 * inv;

    // block-reduce S2
    s_red[tid] = s2; __syncthreads();
    for (int s = 128; s > 0; s >>= 1) { if (tid < s) s_red[tid] += s_red[tid + s]; __syncthreads(); }
    const float s2tot = s_red[0]; __syncthreads();
    // block-reduce ||mus||^2
    s_red[tid] = a0 * a0 + a1 * a1 + a2 * a2 + a3 * a3; __syncthreads();
    for (int s = 128; s > 0; s >>= 1) { if (tid < s) s_red[tid] += s_red[tid + s]; __syncthreads(); }
    if (tid == 0) {
        const float mu2 = s_red[0];
        cls_cnt[k] = cntf;
        cls_s2[k]  = s2tot;
        cls_mu2[k] = mu2;
        cls_pn2[k] = mu2 * inv * inv;
    }
}

// ---------------------------------------------------------------------------
// Kernel B: per-query scalars. One block per query i (class c = ys[pos[i]]):
//   q2 = ||xq_i||^2, qmu = xq_i . mus_c, qxp = xq_i . xs_pos, xp2 = ||xs_pos||^2
// ---------------------------------------------------------------------------
__global__ void __launch_bounds__(256)
query_stats_kernel(const float* __restrict__ xq, const float* __restrict__ xs,
                   const int* __restrict__ ys, const int* __restrict__ pos,
                   const float* __restrict__ mus,
                   float* __restrict__ q2o, float* __restrict__ qmuo,
                   float* __restrict__ qxpo, float* __restrict__ xp2o)
{
    __shared__ float r0[256], r1[256], r2[256], r3[256];
    const int i   = blockIdx.x;
    const int tid = threadIdx.x;
    const int p   = pos[i];
    const int c   = ys[p];
    const float* xqr = xq  + (size_t)i * DIM;
    const float* mur = mus + (size_t)c * DIM;
    const float* xpr = xs  + (size_t)p * DIM;
    float a = 0.f, b = 0.f, cc = 0.f, d = 0.f;
    for (int dd = tid; dd < DIM; dd += 256) {
        const float xv = xqr[dd], mv = mur[dd], pv = xpr[dd];
        a  += xv * xv;
        b  += xv * mv;
        cc += xv * pv;
        d  += pv * pv;
    }
    r0[tid] = a; r1[tid] = b; r2[tid] = cc; r3[tid] = d; __syncthreads();
    for (int s = 128; s > 0; s >>= 1) {
        if (tid < s) { r0[tid] += r0[tid+s]; r1[tid] += r1[tid+s]; r2[tid] += r2[tid+s]; r3[tid] += r3[tid+s]; }
        __syncthreads();
    }
    if (tid == 0) { q2o[i] = r0[0]; qmuo[i] = r1[0]; qxpo[i] = r2[0]; xp2o[i] = r3[0]; }
}

// ---------------------------------------------------------------------------
// Kernel C: G = xq [NQ,D] @ proto_n^T [D,K]  via V_WMMA_F32_16X16X4_F32.
// One wave owns one 16x16 output tile; 8 waves/block = all 8 column tiles of
// one 16-row stripe.  A-frag per ISA 7.12.2 (32-bit A 16x4): lanes 0-15 carry
// K=0,1 in VGPR0/1, lanes 16-31 carry K=2,3.  B fed from protoT (row-major
// over D) so half-wave loads are contiguous across classes.
// ---------------------------------------------------------------------------
__global__ void __launch_bounds__(256)
gemm_qproto_kernel(const float* __restrict__ xq, const float* __restrict__ protoT,
                   float* __restrict__ G)
{
    const int lane = threadIdx.x & 31;
    const int wave = threadIdx.x >> 5;        // column tile 0..7
    const int mt   = blockIdx.x;              // row tile 0..255
    const int half = lane >> 4;               // 0: K{0,1} / M rows, 1: K{2,3}
    const int l15  = lane & 15;
    const int arow = mt * 16 + l15;
    const int bcol = wave * 16 + l15;

    const float* aptr = xq + (size_t)arow * DIM + half * 2;
    v8f acc = {};
    #pragma unroll 4
    for (int k = 0; k < DIM; k += 4) {
        __builtin_prefetch(aptr + k + 64, 0, 0);   // global_prefetch_b8
        v2f a, b;
        a.x = aptr[k];
        a.y = aptr[k + 1];
        const float* bp = protoT + (size_t)(k + half * 2) * KCLS + bcol;
        b.x = bp[0];
        b.y = bp[KCLS];
        acc = __builtin_amdgcn_wmma_f32_16x16x4_f32(
            /*neg_a=*/false, a, /*neg_b=*/false, b,
            /*c_mod=*/(short)0, acc, /*reuse_a=*/false, /*reuse_b=*/false);
    }
    // C/D layout: VGPR r holds row r (lanes 0-15) and row r+8 (lanes 16-31).
    #pragma unroll
    for (int r = 0; r < 8; ++r) {
        const int row = mt * 16 + half * 8 + r;
        G[(size_t)row * KCLS + bcol] = acc[r];
    }
}

// ---------------------------------------------------------------------------
// Kernel D: per-query logsumexp over 128 prototype logits + closed-form
// positive logit; atomic mean into out.  One wave per query.
// (setup guarantees yq[i] == ys[pos[i]], so one class id serves both roles.)
// ---------------------------------------------------------------------------
__global__ void __launch_bounds__(256)
loss_kernel(const int* __restrict__ ys, const int* __restrict__ pos,
            const float* __restrict__ G,
            const float* __restrict__ q2a, const float* __restrict__ qmua,
            const float* __restrict__ qxpa, const float* __restrict__ xp2a,
            const float* __restrict__ cls_cnt, const float* __restrict__ cls_s2,
            const float* __restrict__ cls_mu2, const float* __restrict__ cls_pn2,
            float* __restrict__ out)
{
    const int lane = threadIdx.x & 31;
    const int i    = blockIdx.x * 8 + (threadIdx.x >> 5);
    const int p    = pos[i];
    const int c    = ys[p];

    const float q2  = q2a[i];
    const float qmu = qmua[i];
    const float n   = cls_cnt[c];

    // leave-one-out prototype distance:
    //   M = mus_c - xq ; proto = M / C ; d_pos = sqrt(||xq - proto||^2 + EPS)
    const float C   = fmaxf(n - 1.0f, 0.1f);
    const float m2  = cls_mu2[c] - 2.0f * qmu + q2;   // ||M||^2
    const float qM  = qmu - q2;                       // xq . M
    const float dd  = q2 - 2.0f * qM / C + m2 / (C * C);
    const float d_pos = sqrtf(dd + EPS_);

    // denominator logits: 4 classes per lane
    const float* Gi = G + (size_t)i * KCLS;
    float v[4];
    #pragma unroll
    for (int t = 0; t < 4; ++t) {
        const int k = lane + 32 * t;
        const float sq = fmaxf(q2 + cls_pn2[k] - 2.0f * Gi[k], 0.0f);
        float lg = -sqrtf(sq + EPS_);
        if (k == c) lg = -d_pos;
        v[t] = lg;
    }
    float m = fmaxf(fmaxf(v[0], v[1]), fmaxf(v[2], v[3]));
    #pragma unroll
    for (int s = 16; s > 0; s >>= 1) m = fmaxf(m, __shfl_xor(m, s, 32));
    float se = 0.0f;
    #pragma unroll
    for (int t = 0; t < 4; ++t) se += __expf(v[t] - m);
    #pragma unroll
    for (int s = 16; s > 0; s >>= 1) se += __shfl_xor(se, s, 32);
    const float neg_logit = m + __logf(se);

    if (lane == 0) {
        // masked average of -||xq - xs_j||^2 / 2 over same-class supports:
        //   sum_j!=pos ||xq-xs_j||^2 = n*q2 + S2_c - 2*qmu - ||xq - xs_pos||^2
        float pos_logit = 0.0f;
        if (n > 1.5f) {
            const float sqp = fmaxf(q2 + xp2a[i] - 2.0f * qxpa[i], 0.0f);
            const float classSq = n * q2 + cls_s2[c] - 2.0f * qmu - sqp;
            pos_logit = -0.5f * classSq / (n - 1.0f);
        }
        atomicAdd(out, (neg_logit - pos_logit) * (1.0f / (float)NQ));
    }
}

// ---------------------------------------------------------------------------
extern "C" void kernel_launch(void* const* d_in, const int* in_sizes, int n_in,
                              void* d_out, int out_size, void* d_ws, size_t ws_size,
                              hipStream_t stream) {
    (void)in_sizes; (void)n_in; (void)out_size; (void)ws_size;
    const float* xq  = (const float*)d_in[0];
    const int*   yq  = (const int*)  d_in[1];  (void)yq; // == ys[pos] by setup
    const float* xs  = (const float*)d_in[2];
    const int*   ys  = (const int*)  d_in[3];
    const int*   pos = (const int*)  d_in[4];
    float* out = (float*)d_out;

    // workspace carve-up (floats)
    float* ws      = (float*)d_ws;
    float* mus     = ws;                         // K*D      = 131072
    float* protoT  = mus    + KCLS * DIM;        // D*K      = 131072
    float* G       = protoT + DIM * KCLS;        // NQ*K     = 524288
    float* q2a     = G      + (size_t)NQ * KCLS; // NQ
    float* qmua    = q2a  + NQ;
    float* qxpa    = qmua + NQ;
    float* xp2a    = qxpa + NQ;
    float* cls_cnt = xp2a + NQ;                  // K each
    float* cls_s2  = cls_cnt + KCLS;
    float* cls_mu2 = cls_s2  + KCLS;
    float* cls_pn2 = cls_mu2 + KCLS;

    zero_out_kernel<<<1, 1, 0, stream>>>(out);
    class_stats_kernel<<<KCLS, 256, 0, stream>>>(xs, ys, mus, protoT,
                                                 cls_cnt, cls_s2, cls_mu2, cls_pn2);
    query_stats_kernel<<<NQ, 256, 0, stream>>>(xq, xs, ys, pos, mus,
                                               q2a, qmua, qxpa, xp2a);
    gemm_qproto_kernel<<<NQ / 16, 256, 0, stream>>>(xq, protoT, G);
    loss_kernel<<<NQ / 8, 256, 0, stream>>>(ys, pos, G, q2a, qmua, qxpa, xp2a,
                                            cls_cnt, cls_s2, cls_mu2, cls_pn2, out);
}